// Optimal_Transport_Layer_3779571221175
// MI455X (gfx1250) — compile-verified
//
#include <hip/hip_runtime.h>
#include <hip/hip_bf16.h>
#include <math.h>

#define NPTS 4096
#define N1   4097
#define FDIM 128
#define NORM_C   (-9.0109133472f)   /* -log(8192) */
#define BIN_LMU  (-0.6931471806f)   /* log(4096) - log(8192) = -ln2 */
#define ITERS 100

typedef __attribute__((ext_vector_type(2))) float v2f;
typedef __attribute__((ext_vector_type(8))) float v8f;

// ---------------------------------------------------------------------------
// 1) GEMM: C[i][j] = (1/sqrt(128)) * sum_k mdesc0[k][i] * mdesc1[k][j]
//    One wave per 16x16 tile, V_WMMA_F32_16X16X4_F32, K-loop of 32 steps.
//    A-frag layout (ISA 7.12.2, 32-bit A 16x4): lanes 0-15 hold K={0,1},
//    lanes 16-31 hold K={2,3}; B symmetric. C/D: vgpr v = row (v + 8*half).
// ---------------------------------------------------------------------------
__global__ __launch_bounds__(256) void ot_gemm_wmma(const float* __restrict__ A,
                                                    const float* __restrict__ B,
                                                    float* __restrict__ C) {
  const int lane = threadIdx.x & 31;
  const int wave = threadIdx.x >> 5;
  const int half = lane >> 4;
  const int l    = lane & 15;
  const int i0 = (blockIdx.x * 8 + wave) * 16;
  const int j0 = blockIdx.y * 16;

  v8f acc = {};
#pragma unroll
  for (int k0 = 0; k0 < FDIM; k0 += 4) {
    const int kb = k0 + 2 * half;
    v2f a, b;
    a.x = A[(size_t)kb       * NPTS + i0 + l];
    a.y = A[(size_t)(kb + 1) * NPTS + i0 + l];
    b.x = B[(size_t)kb       * NPTS + j0 + l];
    b.y = B[(size_t)(kb + 1) * NPTS + j0 + l];
    acc = __builtin_amdgcn_wmma_f32_16x16x4_f32(false, a, false, b,
                                                (short)0, acc, false, false);
  }
  const float scale = 0.08838834764831845f; // 1/sqrt(128)
#pragma unroll
  for (int v = 0; v < 8; ++v) {
    const int row = v + half * 8;
    C[(size_t)(i0 + row) * N1 + (j0 + l)] = acc[v] * scale;
  }
}

// ---------------------------------------------------------------------------
// 2) Fill dustbin row/col/corner with alpha; zero log_u / log_v.
// ---------------------------------------------------------------------------
__global__ void ot_dustbin_init(float* __restrict__ C, const float* __restrict__ bin,
                                float* __restrict__ log_u, float* __restrict__ log_v) {
  const int t = blockIdx.x * blockDim.x + threadIdx.x;
  const float alpha = *bin;
  if (t < NPTS) {
    C[(size_t)t * N1 + NPTS] = alpha;      // bins0 column
    C[(size_t)NPTS * N1 + t] = alpha;      // bins1 row
  } else if (t == NPTS) {
    C[(size_t)NPTS * N1 + NPTS] = alpha;   // corner
  }
  if (t < N1) { log_u[t] = 0.0f; log_v[t] = 0.0f; }
}

// ---------------------------------------------------------------------------
// 3) Row sweep: log_u[i] = log_mu[i] - logsumexp_j(C[i][j] + log_v[j])
//    One block per row; online (m,s) per thread, LDS tree combine.
// ---------------------------------------------------------------------------
__global__ __launch_bounds__(256) void ot_row_pass(const float* __restrict__ C,
                                                   const float* __restrict__ log_v,
                                                   float* __restrict__ log_u) {
  __shared__ float sm[256];
  __shared__ float ss[256];
  const int i = blockIdx.x;
  const size_t rb = (size_t)i * N1;
  float m = -INFINITY, s = 0.0f;
  for (int j = threadIdx.x; j < N1; j += 256) {
    const float x = C[rb + j] + log_v[j];
    if (x > m) { s = s * __expf(m - x) + 1.0f; m = x; }
    else       { s += __expf(x - m); }
  }
  sm[threadIdx.x] = m; ss[threadIdx.x] = s;
  __syncthreads();
  for (int off = 128; off > 0; off >>= 1) {
    if (threadIdx.x < off) {
      const float m1 = sm[threadIdx.x], s1 = ss[threadIdx.x];
      const float m2 = sm[threadIdx.x + off], s2 = ss[threadIdx.x + off];
      const float M = fmaxf(m1, m2);
      sm[threadIdx.x] = M;
      ss[threadIdx.x] = s1 * __expf(m1 - M) + s2 * __expf(m2 - M);
    }
    __syncthreads();
  }
  if (threadIdx.x == 0) {
    const float lse = sm[0] + __logf(ss[0]);
    const float lmu = (i < NPTS) ? NORM_C : BIN_LMU;
    log_u[i] = lmu - lse;
  }
}

// ---------------------------------------------------------------------------
// 4) Column sweep: log_v[j] = log_nu[j] - logsumexp_i(C[i][j] + log_u[i])
//    Block owns 32 columns x 8 row-splits (coalesced along j), LDS combine.
// ---------------------------------------------------------------------------
__global__ __launch_bounds__(256) void ot_col_pass(const float* __restrict__ C,
                                                   const float* __restrict__ log_u,
                                                   float* __restrict__ log_v) {
  __shared__ float sm[8][32];
  __shared__ float ss[8][32];
  const int jl = threadIdx.x & 31;
  const int r  = threadIdx.x >> 5;      // 0..7 row split
  const int j  = blockIdx.x * 32 + jl;
  const int i_begin = r * 513;
  const int i_end   = (i_begin + 513 < N1) ? (i_begin + 513) : N1;
  float m = -INFINITY, s = 0.0f;
  if (j < N1) {
    for (int i = i_begin; i < i_end; ++i) {
      const float x = C[(size_t)i * N1 + j] + log_u[i];
      if (x > m) { s = s * __expf(m - x) + 1.0f; m = x; }
      else       { s += __expf(x - m); }
    }
  }
  sm[r][jl] = m; ss[r][jl] = s;
  __syncthreads();
  if (r == 0 && j < N1) {
    float M = sm[0][jl], S = ss[0][jl];
#pragma unroll
    for (int q = 1; q < 8; ++q) {
      const float m2 = sm[q][jl], s2 = ss[q][jl];
      const float Mn = fmaxf(M, m2);
      S = S * __expf(M - Mn) + s2 * __expf(m2 - Mn);
      M = Mn;
    }
    const float lse = M + __logf(S);
    const float lnu = (j < NPTS) ? NORM_C : BIN_LMU;
    log_v[j] = lnu - lse;
  }
}

// ---------------------------------------------------------------------------
// 5) Finalize rows: E = exp(C + log_u + log_v - norm) written IN PLACE,
//    plus row max/argmax over inner region (j < 4096) for i < 4096.
// ---------------------------------------------------------------------------
__global__ __launch_bounds__(256) void ot_finalize_rows(float* __restrict__ C,
                                                        const float* __restrict__ log_u,
                                                        const float* __restrict__ log_v,
                                                        float* __restrict__ rowmax,
                                                        int* __restrict__ idx0) {
  __shared__ float sv[256];
  __shared__ int   si[256];
  const int i = blockIdx.x;
  const size_t rb = (size_t)i * N1;
  const float lu = log_u[i];
  float bestv = -1.0f; int bestj = 0;
  for (int j = threadIdx.x; j < N1; j += 256) {
    const float z = C[rb + j] + lu + log_v[j] - NORM_C;
    const float e = __expf(z);
    C[rb + j] = e;
    if (j < NPTS && e > bestv) { bestv = e; bestj = j; }
  }
  sv[threadIdx.x] = bestv; si[threadIdx.x] = bestj;
  __syncthreads();
  for (int off = 128; off > 0; off >>= 1) {
    if (threadIdx.x < off) {
      const float v2 = sv[threadIdx.x + off]; const int j2 = si[threadIdx.x + off];
      if (v2 > sv[threadIdx.x] ||
          (v2 == sv[threadIdx.x] && j2 < si[threadIdx.x])) {
        sv[threadIdx.x] = v2; si[threadIdx.x] = j2;
      }
    }
    __syncthreads();
  }
  if (threadIdx.x == 0 && i < NPTS) { rowmax[i] = sv[0]; idx0[i] = si[0]; }
}

// ---------------------------------------------------------------------------
// 6) Column argmax over inner region (i < 4096) of exp matrix.
// ---------------------------------------------------------------------------
__global__ __launch_bounds__(256) void ot_col_argmax(const float* __restrict__ E,
                                                     int* __restrict__ idx1) {
  __shared__ float sv[8][32];
  __shared__ int   si[8][32];
  const int jl = threadIdx.x & 31;
  const int r  = threadIdx.x >> 5;
  const int j  = blockIdx.x * 32 + jl;   // grid = 128 -> j < 4096 always
  float bestv = -1.0f; int besti = 0;
  const int i_begin = r * 512;
  for (int i = i_begin; i < i_begin + 512; ++i) {
    const float e = E[(size_t)i * N1 + j];
    if (e > bestv) { bestv = e; besti = i; }
  }
  sv[r][jl] = bestv; si[r][jl] = besti;
  __syncthreads();
  if (r == 0) {
#pragma unroll
    for (int q = 1; q < 8; ++q) {
      if (sv[q][jl] > bestv || (sv[q][jl] == bestv && si[q][jl] < besti)) {
        bestv = sv[q][jl]; besti = si[q][jl];
      }
    }
    idx1[j] = besti;
  }
}

// ---------------------------------------------------------------------------
// 7) Mutual matching, side 0 then side 1.
// ---------------------------------------------------------------------------
__global__ void ot_match0(const float* __restrict__ rowmax, const int* __restrict__ idx0,
                          const int* __restrict__ idx1,
                          float* __restrict__ outI0, float* __restrict__ outM0,
                          int* __restrict__ valid0, float* __restrict__ msc0) {
  const int i = blockIdx.x * blockDim.x + threadIdx.x;
  if (i >= NPTS) return;
  const int j = idx0[i];
  const bool mutual = (idx1[j] == i);
  const float ms = mutual ? rowmax[i] : 0.0f;     // rowmax already = exp(max Z)
  const bool v = mutual && (ms > 0.2f);
  outM0[i] = ms;
  msc0[i]  = ms;
  valid0[i] = v ? 1 : 0;
  outI0[i] = v ? (float)j : -1.0f;
}

__global__ void ot_match1(const int* __restrict__ idx0, const int* __restrict__ idx1,
                          const float* __restrict__ msc0, const int* __restrict__ valid0,
                          float* __restrict__ outI1, float* __restrict__ outM1) {
  const int j = blockIdx.x * blockDim.x + threadIdx.x;
  if (j >= NPTS) return;
  const int i = idx1[j];
  const bool mutual = (idx0[i] == j);
  const float ms = mutual ? msc0[i] : 0.0f;
  const bool v = mutual && (valid0[i] != 0);
  outM1[j] = ms;
  outI1[j] = v ? (float)i : -1.0f;
}

// ---------------------------------------------------------------------------
extern "C" void kernel_launch(void* const* d_in, const int* in_sizes, int n_in,
                              void* d_out, int out_size, void* d_ws, size_t ws_size,
                              hipStream_t stream) {
  const float* mdesc0 = (const float*)d_in[0];   // (128, 4096)
  const float* mdesc1 = (const float*)d_in[1];   // (128, 4096)
  const float* bin    = (const float*)d_in[2];   // scalar

  float* out   = (float*)d_out;
  float* C     = out;                              // 4097*4097, couplings -> exp(Z)
  float* outI0 = out + (size_t)N1 * N1;
  float* outI1 = outI0 + NPTS;
  float* outM0 = outI1 + NPTS;
  float* outM1 = outM0 + NPTS;

  float* ws     = (float*)d_ws;
  float* log_u  = ws;                // 4097 (padded slots of 4128)
  float* log_v  = ws + 4128;
  float* rowmax = ws + 2 * 4128;
  int*   idx0   = (int*)(ws + 3 * 4128);
  int*   idx1   = (int*)(ws + 4 * 4128);
  int*   valid0 = (int*)(ws + 5 * 4128);
  float* msc0   = ws + 6 * 4128;

  // 1) scores GEMM via f32 WMMA (one wave per 16x16 tile, 8 waves/block)
  dim3 gGemm(NPTS / (16 * 8), NPTS / 16);          // (32, 256)
  ot_gemm_wmma<<<gGemm, 256, 0, stream>>>(mdesc0, mdesc1, C);

  // 2) dustbins + zero log_u/log_v
  ot_dustbin_init<<<(N1 + 255) / 256, 256, 0, stream>>>(C, bin, log_u, log_v);

  // 3) Sinkhorn iterations (L2-resident 67MB matrix, 200 sweeps)
  for (int it = 0; it < ITERS; ++it) {
    ot_row_pass<<<N1, 256, 0, stream>>>(C, log_v, log_u);
    ot_col_pass<<<(N1 + 31) / 32, 256, 0, stream>>>(C, log_u, log_v);
  }

  // 4) exp(Z) in place + row argmax, then column argmax
  ot_finalize_rows<<<N1, 256, 0, stream>>>(C, log_u, log_v, rowmax, idx0);
  ot_col_argmax<<<NPTS / 32, 256, 0, stream>>>(C, idx1);

  // 5) mutual matching
  ot_match0<<<NPTS / 256, 256, 0, stream>>>(rowmax, idx0, idx1, outI0, outM0, valid0, msc0);
  ot_match1<<<NPTS / 256, 256, 0, stream>>>(idx0, idx1, msc0, valid0, outI1, outM1);
}